// TasteGNN_75179107549407
// MI455X (gfx1250) — compile-verified
//
#include <hip/hip_runtime.h>
#include <cstdint>

typedef float v2f __attribute__((ext_vector_type(2)));
typedef float v8f __attribute__((ext_vector_type(8)));

#define HID  128
#define XPAD 132   // padded LDS row stride: row*132 % 64 banks spreads M-strided reads

// ---------- helpers ----------
__device__ __forceinline__ float atomic_max_float(float* addr, float val) {
  // IEEE ordered-int trick; addr initialized to -inf
  if (val >= 0.f)
    return __int_as_float(atomicMax((int*)addr, __float_as_int(val)));
  else
    return __uint_as_float(atomicMin((unsigned int*)addr, __float_as_uint(val)));
}

// ---------- init (harness poisons buffers; zero/seed everything we accumulate into) ----------
__global__ void k_init(float* __restrict__ agg, float* __restrict__ m,
                       float* __restrict__ denom, float* __restrict__ bnsum,
                       float* __restrict__ bnssq, int NT, int total) {
  int i = blockIdx.x * blockDim.x + threadIdx.x;
  if (i < total) agg[i] = 0.f;
  if (i < NT) { m[i] = -__builtin_inff(); denom[i] = 0.f; }
  if (i < HID) { bnsum[i] = 0.f; bnssq[i] = 0.f; }
}

// ---------- h_src = X @ W^T + b  via v_wmma_f32_16x16x4_f32 ----------
// Block: 256 threads (8 wave32s). Block computes 16 rows x 128 cols of H.
// Wave w owns output columns [16w, 16w+16). K-loop: 32 WMMA steps of K=4.
__global__ __launch_bounds__(256) void k_proj_wmma(
    const float* __restrict__ X, const float* __restrict__ W,
    const float* __restrict__ bias, float* __restrict__ H, int M) {
  __shared__ float xs[16 * XPAD];

  const int row0 = blockIdx.x * 16;
  const int tid  = threadIdx.x;
  const int wave = tid >> 5;
  const int lane = tid & 31;
  const int l16  = lane & 15;
  const int hi   = lane >> 4;          // 0: K pair {0,1}, 1: K pair {2,3}
  const int nb   = wave * 16;          // this wave's N base

  // cooperative coalesced load of the 16x128 x-tile
  for (int f = tid; f < 16 * HID; f += 256) {
    int r = f >> 7, c = f & (HID - 1);
    int rr = row0 + r; if (rr >= M) rr = M - 1;          // clamp (M%16==0 normally)
    xs[r * XPAD + c] = X[(size_t)rr * HID + c];
  }
  __syncthreads();

  v8f acc = {};
  #pragma unroll
  for (int k0 = 0; k0 < HID; k0 += 4) {
    const int kk = k0 + 2 * hi;
    v2f a, b;
    // A (16x4 f32): lane l<16 -> A[M=l, K=k0..k0+1]; lane>=16 -> K=k0+2..k0+3
    a.x = xs[l16 * XPAD + kk];
    a.y = xs[l16 * XPAD + kk + 1];
    // B (4x16 f32): B[k, n] = W[nb+n, k]
    b.x = W[(size_t)(nb + l16) * HID + kk];
    b.y = W[(size_t)(nb + l16) * HID + kk + 1];
    acc = __builtin_amdgcn_wmma_f32_16x16x4_f32(false, a, false, b,
                                                (short)0, acc, false, false);
  }

  const float bv = bias[nb + l16];
  // C/D layout: VGPR v -> lanes 0-15: M=v, N=lane; lanes 16-31: M=v+8, N=lane-16
  #pragma unroll
  for (int v = 0; v < 8; ++v) {
    int mr = row0 + v + 8 * hi;
    if (mr < M) H[(size_t)mr * HID + nb + l16] = acc[v] + bv;
  }
}

// ---------- wave-per-row dot:  out[i] = rows[i] . vec (+ addc[0]) ----------
__global__ void k_rowdot(const float* __restrict__ rows, const float* __restrict__ vec,
                         const float* __restrict__ addc, float* __restrict__ out, int N) {
  int gw   = (int)(((size_t)blockIdx.x * blockDim.x + threadIdx.x) >> 5);
  int lane = threadIdx.x & 31;
  if (gw >= N) return;
  const float* r = rows + (size_t)gw * HID;
  float s = 0.f;
  #pragma unroll
  for (int c = lane; c < HID; c += 32) s += r[c] * vec[c];
  #pragma unroll
  for (int off = 16; off > 0; off >>= 1) s += __shfl_down(s, off, 32);
  if (lane == 0) out[gw] = s + (addc ? addc[0] : 0.f);
}

// ---------- u[k] = sum_d W_taste[d,k]*att_dst[d];  c = b_taste . att_dst ----------
__global__ void k_make_u(const float* __restrict__ Wt, const float* __restrict__ att,
                         const float* __restrict__ bt, float* __restrict__ u,
                         float* __restrict__ c) {
  int k = threadIdx.x;                 // 128 threads
  float s = 0.f;
  for (int d = 0; d < HID; ++d) s += Wt[(size_t)d * HID + k] * att[d];
  u[k] = s;
  if (k == 0) {
    float cc = 0.f;
    for (int d = 0; d < HID; ++d) cc += bt[d] * att[d];
    *c = cc;
  }
}

// ---------- edge pass 1: segment max of leaky-relu logits ----------
__global__ void k_edge_max(const long long* __restrict__ src, const long long* __restrict__ dst,
                           const float* __restrict__ as, const float* __restrict__ ad,
                           float* __restrict__ m, int E) {
  int e = blockIdx.x * blockDim.x + threadIdx.x;
  if (e >= E) return;
  int s = (int)src[e], d = (int)dst[e];
  float a = as[s] + ad[d];
  a = (a > 0.f) ? a : 0.2f * a;
  atomic_max_float(&m[d], a);
}

// ---------- edge pass 2: e = exp(alpha - m[dst]); denom[dst] += e ----------
__global__ void k_edge_exp(const long long* __restrict__ src, const long long* __restrict__ dst,
                           const float* __restrict__ as, const float* __restrict__ ad,
                           const float* __restrict__ m, float* __restrict__ ebuf,
                           float* __restrict__ denom, int E) {
  int e = blockIdx.x * blockDim.x + threadIdx.x;
  if (e >= E) return;
  int s = (int)src[e], d = (int)dst[e];
  float a = as[s] + ad[d];
  a = (a > 0.f) ? a : 0.2f * a;
  float ex = expf(a - m[d]);
  ebuf[e] = ex;
  atomicAdd(&denom[d], ex);
}

// ---------- edge pass 3: agg[dst] += coef * h_src[src]  (wave per edge, float4/lane) ----------
__global__ void k_edge_scatter(const long long* __restrict__ src, const long long* __restrict__ dst,
                               const float* __restrict__ ebuf, const float* __restrict__ denom,
                               const float* __restrict__ H, float* __restrict__ agg, int E) {
  int gw   = (int)(((size_t)blockIdx.x * blockDim.x + threadIdx.x) >> 5);
  int lane = threadIdx.x & 31;
  if (gw >= E) return;
  int s = (int)src[gw], d = (int)dst[gw];
  float dn = denom[d];
  float coef = ebuf[gw] / (dn == 0.f ? 1.f : dn);
  const float4 hv = ((const float4*)(H + (size_t)s * HID))[lane];
  float* ar = agg + (size_t)d * HID + lane * 4;
  atomicAdd(ar + 0, coef * hv.x);
  atomicAdd(ar + 1, coef * hv.y);
  atomicAdd(ar + 2, coef * hv.z);
  atomicAdd(ar + 3, coef * hv.w);
}

// ---------- BN stats over relu(agg): per-feature sum / sumsq ----------
__global__ __launch_bounds__(256) void k_bn_stats(const float* __restrict__ agg,
                                                  float* __restrict__ sum,
                                                  float* __restrict__ ssq, int N) {
  __shared__ float sh[256];
  int c    = threadIdx.x & (HID - 1);
  int half = threadIdx.x >> 7;         // 0 or 1
  float s = 0.f, q = 0.f;
  for (int r = blockIdx.x * 2 + half; r < N; r += gridDim.x * 2) {
    float v = agg[(size_t)r * HID + c];
    v = v > 0.f ? v : 0.f;             // ReLU before BN (as in reference)
    s += v; q += v * v;
  }
  sh[threadIdx.x] = s; __syncthreads();
  if (half == 0) atomicAdd(&sum[c], s + sh[c + HID]);
  __syncthreads();
  sh[threadIdx.x] = q; __syncthreads();
  if (half == 0) atomicAdd(&ssq[c], q + sh[c + HID]);
}

// ---------- BN normalize + final ReLU (in place on d_out) ----------
__global__ void k_bn_norm(float* __restrict__ out, const float* __restrict__ sum,
                          const float* __restrict__ ssq, const float* __restrict__ gamma,
                          const float* __restrict__ beta, int N) {
  int i = blockIdx.x * blockDim.x + threadIdx.x;
  int total = N * HID;
  if (i >= total) return;
  int c = i & (HID - 1);
  float inv_n = 1.f / (float)N;
  float mu  = sum[c] * inv_n;
  float var = ssq[c] * inv_n - mu * mu;        // population variance (ddof=0)
  float v = out[i]; v = v > 0.f ? v : 0.f;     // ReLU (pre-BN, matches stats pass)
  v = (v - mu) * rsqrtf(var + 1e-5f) * gamma[c] + beta[c];
  out[i] = v > 0.f ? v : 0.f;                  // final ReLU
}

// ---------- launch ----------
extern "C" void kernel_launch(void* const* d_in, const int* in_sizes, int n_in,
                              void* d_out, int out_size, void* d_ws, size_t ws_size,
                              hipStream_t stream) {
  const float*     x_ing  = (const float*)d_in[0];
  const float*     x_ta   = (const float*)d_in[1];
  const long long* eidx   = (const long long*)d_in[2];   // int64 [2, E]
  const float*     W_ing  = (const float*)d_in[3];
  const float*     b_ing  = (const float*)d_in[4];
  const float*     W_ta   = (const float*)d_in[5];
  const float*     b_ta   = (const float*)d_in[6];
  const float*     att_s  = (const float*)d_in[7];
  const float*     att_d  = (const float*)d_in[8];
  // d_in[9..11] (k_lin_W, k_lin_b, q) unused: softmax over one metapath == 1.0 exactly
  const float*     gamma  = (const float*)d_in[12];
  const float*     beta   = (const float*)d_in[13];

  const int NI = in_sizes[0] / HID;
  const int NT = in_sizes[1] / HID;
  const int E  = in_sizes[2] / 2;
  const long long* esrc = eidx;
  const long long* edst = eidx + E;

  float* agg = (float*)d_out;                  // NT x 128, doubles as agg buffer

  float* ws     = (float*)d_ws;
  float* h_src  = ws;                          // NI*128
  float* a_src  = h_src + (size_t)NI * HID;    // NI
  float* a_dst  = a_src + NI;                  // NT
  float* m      = a_dst + NT;                  // NT
  float* denom  = m + NT;                      // NT
  float* ebuf   = denom + NT;                  // E
  float* u      = ebuf + E;                    // 128
  float* cbuf   = u + HID;                     // 1
  float* bnsum  = cbuf + 1;                    // 128
  float* bnssq  = bnsum + HID;                 // 128

  const int total = NT * HID;

  k_init<<<(total + 255) / 256, 256, 0, stream>>>(agg, m, denom, bnsum, bnssq, NT, total);

  k_proj_wmma<<<(NI + 15) / 16, 256, 0, stream>>>(x_ing, W_ing, b_ing, h_src, NI);

  k_rowdot<<<(NI + 7) / 8, 256, 0, stream>>>(h_src, att_s, nullptr, a_src, NI);

  k_make_u<<<1, HID, 0, stream>>>(W_ta, att_d, b_ta, u, cbuf);

  k_rowdot<<<(NT + 7) / 8, 256, 0, stream>>>(x_ta, u, cbuf, a_dst, NT);

  k_edge_max<<<(E + 255) / 256, 256, 0, stream>>>(esrc, edst, a_src, a_dst, m, E);

  k_edge_exp<<<(E + 255) / 256, 256, 0, stream>>>(esrc, edst, a_src, a_dst, m, ebuf, denom, E);

  k_edge_scatter<<<(E + 7) / 8, 256, 0, stream>>>(esrc, edst, ebuf, denom, h_src, agg, E);

  k_bn_stats<<<256, 256, 0, stream>>>(agg, bnsum, bnssq, NT);

  k_bn_norm<<<(total + 255) / 256, 256, 0, stream>>>(agg, bnsum, bnssq, gamma, beta, NT);
}